// RWKV_LAYER_56822417326424
// MI455X (gfx1250) — compile-verified
//
#include <hip/hip_runtime.h>
#include <math.h>

#define E 1024
#define LAYERS 24
#define FF 4096
#define NWG 128
#define NT 256
#define NWAVE 8

typedef __attribute__((ext_vector_type(2))) float v2f;
typedef __attribute__((ext_vector_type(8))) float v8f;

// workspace layout (float offsets)
#define WS_XCUR 0
#define WS_SX   1024
#define WS_K    2048
#define WS_V    3072
#define WS_R    4096
#define WS_KF   5120
#define WS_DEN  9216
#define WS_BAR  10240   // 2 unsigned

#if __has_builtin(__builtin_amdgcn_wmma_f32_16x16x4_f32)
#define WMMA_F32_4(acc, a, b) \
  acc = __builtin_amdgcn_wmma_f32_16x16x4_f32(false, a, false, b, (short)0, acc, false, false)
#else
#define WMMA_F32_4(acc, a, b) \
  asm volatile("v_wmma_f32_16x16x4_f32 %0, %1, %2, %0" : "+v"(acc) : "v"(a), "v"(b))
#endif

// -------- device-wide barrier (sense reversing) --------
__device__ __forceinline__ void grid_barrier(unsigned* bar) {
  __syncthreads();
  if (threadIdx.x == 0) {
    __threadfence();                                  // release my WG's writes to device scope
    unsigned g = ((volatile unsigned*)bar)[1];        // read generation BEFORE arriving
    unsigned a = atomicAdd(&bar[0], 1u);
    if (a == NWG - 1u) {
      ((volatile unsigned*)bar)[0] = 0u;
      __threadfence();
      atomicExch(&bar[1], g + 1u);                    // flip generation
    } else {
      while (((volatile unsigned*)bar)[1] == g) __builtin_amdgcn_s_sleep(1);
      __threadfence();                                // acquire: invalidate near caches
    }
  }
  __syncthreads();
}

// -------- block mean / rstd over a 1024-vector in global memory --------
__device__ __forceinline__ void block_meanrstd(const float* __restrict__ v, float* stat,
                                               float& mu, float& rstd) {
  const int tid = threadIdx.x, lane = tid & 31, wave = tid >> 5;
  float s1 = 0.f, s2 = 0.f;
  for (int i = tid; i < E; i += NT) { float x = v[i]; s1 += x; s2 += x * x; }
#pragma unroll
  for (int off = 16; off > 0; off >>= 1) {
    s1 += __shfl_xor(s1, off, 32);
    s2 += __shfl_xor(s2, off, 32);
  }
  if (lane == 0) { stat[wave] = s1; stat[8 + wave] = s2; }
  __syncthreads();
  if (tid == 0) {
    float a = 0.f, c = 0.f;
    for (int w = 0; w < NWAVE; ++w) { a += stat[w]; c += stat[8 + w]; }
    float m = a * (1.0f / E);
    stat[16] = m;
    stat[17] = rsqrtf(c * (1.0f / E) - m * m + 1e-5f);
  }
  __syncthreads();
  mu = stat[16];
  rstd = stat[17];
}

// -------- 16-row GEMV tile via V_WMMA_F32_16X16X4_F32 --------
// dot16[j] = W[j, 0:K] . B[0:K]   (W row-major, stride ldw; B in LDS)
// 8 waves split K; per-wave partials (D column 0, lanes 0 & 16) reduced via LDS.
__device__ __forceinline__ void gemv16_wmma(const float* __restrict__ W, int ldw, int K,
                                            const float* B, float* redbuf, float* tileres) {
  const int tid = threadIdx.x;
  const int lane = tid & 31;
  const int wave = tid >> 5;
  const int mrow = lane & 15;   // A: lane%16 = M row
  const int khalf = lane >> 4;  // A/B: upper half-wave holds K+2,K+3
  const int kper = K >> 3;      // K / 8 waves
  const float* wp = W + (size_t)mrow * ldw + wave * kper + 2 * khalf;
  const float* bp = B + wave * kper + 2 * khalf;
  v8f acc0 = {0.f, 0.f, 0.f, 0.f, 0.f, 0.f, 0.f, 0.f};
  v8f acc1 = {0.f, 0.f, 0.f, 0.f, 0.f, 0.f, 0.f, 0.f};
  for (int k = 0; k < kper; k += 8) {
    v2f a0 = *(const v2f*)(wp);
    v2f b0 = *(const v2f*)(bp);
    v2f a1 = *(const v2f*)(wp + 4);
    v2f b1 = *(const v2f*)(bp + 4);
    WMMA_F32_4(acc0, a0, b0);
    WMMA_F32_4(acc1, a1, b1);
    wp += 8;
    bp += 8;
  }
  acc0 += acc1;
  // D 16x16 f32: lane 0 VGPR j = (M=j, N=0); lane 16 VGPR j = (M=8+j, N=0)
  if (mrow == 0) {
    float* dst = redbuf + wave * 16 + khalf * 8;
#pragma unroll
    for (int j = 0; j < 8; ++j) dst[j] = acc0[j];
  }
  __syncthreads();
  if (tid < 16) {
    float s = 0.f;
#pragma unroll
    for (int w = 0; w < NWAVE; ++w) s += redbuf[w * 16 + tid];
    tileres[tid] = s;
  }
  __syncthreads();
}

__global__ void rwkv_init_kernel(const float* __restrict__ x, float* __restrict__ ws,
                                 unsigned* __restrict__ bar) {
  int i = blockIdx.x * blockDim.x + threadIdx.x;
  if (i < E) ws[WS_XCUR + i] = x[i];
  if (i < 2) bar[i] = 0u;
}

__global__ __launch_bounds__(NT, 1) void rwkv_fused_kernel(
    const float* __restrict__ statea, const float* __restrict__ stateb,
    const float* __restrict__ statec, const float* __restrict__ stated,
    const float* __restrict__ ln1w, const float* __restrict__ ln1b,
    const float* __restrict__ ln2w, const float* __restrict__ ln2b,
    const float* __restrict__ tdec, const float* __restrict__ tfirst,
    const float* __restrict__ keyw, const float* __restrict__ outw,
    const float* __restrict__ mixk, const float* __restrict__ mixv,
    const float* __restrict__ mixr, const float* __restrict__ mixkf,
    const float* __restrict__ mixrf, const float* __restrict__ keyffn,
    const float* __restrict__ recffn, const float* __restrict__ valffn,
    float* __restrict__ out, float* ws, unsigned* bar) {
  __shared__ float sh_xk[E], sh_xv[E], sh_xr[E], sh_wkv[E];
  __shared__ float sh_kf[FF];
  __shared__ float sh_red[NWAVE * 16];
  __shared__ float sh_tile[16];
  __shared__ float sh_stat[18];

  const int tid = threadIdx.x;
  const int wg = (int)blockIdx.x;

  float* xcur = ws + WS_XCUR;
  float* sxv = ws + WS_SX;
  float* kb = ws + WS_K;
  float* vb = ws + WS_V;
  float* rb = ws + WS_R;
  float* kfv = ws + WS_KF;
  float* denv = ws + WS_DEN;

  float* out_x = out;
  float* out_sa = out + E;
  float* out_sb = out + E + (size_t)LAYERS * E;
  float* out_sc = out + E + (size_t)2 * LAYERS * E;
  float* out_sd = out + E + (size_t)3 * LAYERS * E;

  for (int l = 0; l < LAYERS; ++l) {
    const size_t lE = (size_t)l * E;

    // ---------------- Phase A: LN1 + GEMV k/v/r ----------------
    {
      float mu, rstd;
      block_meanrstd(xcur, sh_stat, mu, rstd);
      for (int i = tid; i < E; i += NT) {
        float xn = (xcur[i] - mu) * rstd * ln1w[lE + i] + ln1b[lE + i];
        float sa = statea[lE + i];
        sh_xk[i] = xn + mixk[lE + i] * sa;
        sh_xv[i] = xn + mixv[lE + i] * sa;
        sh_xr[i] = xn + mixr[lE + i] * sa;
        if (wg == 0) out_sa[lE + i] = xn;
      }
      __syncthreads();
      const float* Wl = keyw + (size_t)l * 3 * E * E;
      for (int t = wg; t < 192; t += NWG) {
        int sub = t >> 6;
        int r0 = (t & 63) << 4;
        const float* B = (sub == 0) ? sh_xk : (sub == 1) ? sh_xv : sh_xr;
        gemv16_wmma(Wl + (size_t)sub * E * E + (size_t)r0 * E, E, E, B, sh_red, sh_tile);
        if (tid < 16) {
          int i = r0 + tid;
          float d = sh_tile[tid];
          if (sub == 0) kb[i] = expf(d);
          else if (sub == 1) vb[i] = d;
          else rb[i] = expf(d) + 1.0f;
        }
      }
      grid_barrier(bar);
    }

    // ---------------- Phase B: WKV + GEMV outputv ----------------
    {
      for (int i = tid; i < E; i += NT) {
        float kk = kb[i], vv = vb[i], rr = rb[i];
        float tf = tfirst[lE + i];
        float sbv = stateb[lE + i];
        float scv = statec[lE + i];
        sh_wkv[i] = (sbv + tf * kk * vv) / (scv * rr + tf * kk * rr);
        if (wg == 0) {
          float td = tdec[lE + i];
          out_sb[lE + i] = sbv * td + kk * vv;
          out_sc[lE + i] = scv * td + kk;
        }
      }
      __syncthreads();
      const float* Wl = outw + (size_t)l * E * E;
      for (int t = wg; t < 64; t += NWG) {
        int r0 = t << 4;
        gemv16_wmma(Wl + (size_t)r0 * E, E, E, sh_wkv, sh_red, sh_tile);
        if (tid < 16) {
          int i = r0 + tid;
          sxv[i] = xcur[i] + sh_tile[tid];
        }
      }
      grid_barrier(bar);
    }

    // ---------------- Phase C: LN2 + GEMV key_ffn & receptance_ffn ----------------
    {
      float mu, rstd;
      block_meanrstd(sxv, sh_stat, mu, rstd);
      for (int i = tid; i < E; i += NT) {
        float x2 = (sxv[i] - mu) * rstd * ln2w[lE + i] + ln2b[lE + i];
        float sd = stated[lE + i];
        sh_xk[i] = x2 + mixkf[lE + i] * sd;
        sh_xr[i] = x2 + mixrf[lE + i] * sd;
        if (wg == 0) out_sd[lE + i] = x2;
      }
      __syncthreads();
      const float* Wk = keyffn + (size_t)l * FF * E;
      const float* Wr = recffn + (size_t)l * E * E;
      for (int t = wg; t < 320; t += NWG) {
        if (t < 256) {
          int r0 = t << 4;
          gemv16_wmma(Wk + (size_t)r0 * E, E, E, sh_xk, sh_red, sh_tile);
          if (tid < 16) {
            int i = r0 + tid;
            float rl = fmaxf(sh_tile[tid], 0.0f);
            kfv[i] = rl * rl;
          }
        } else {
          int r0 = (t - 256) << 4;
          gemv16_wmma(Wr + (size_t)r0 * E, E, E, sh_xr, sh_red, sh_tile);
          if (tid < 16) {
            int i = r0 + tid;
            denv[i] = 1.0f / (1.0f + expf(sh_tile[tid]));
          }
        }
      }
      grid_barrier(bar);
    }

    // ---------------- Phase D: GEMV value_ffn, write next x ----------------
    {
      for (int i = tid; i < FF; i += NT) sh_kf[i] = kfv[i];
      __syncthreads();
      const float* Wv = valffn + (size_t)l * E * FF;
      for (int t = wg; t < 64; t += NWG) {
        int r0 = t << 4;
        gemv16_wmma(Wv + (size_t)r0 * FF, FF, FF, sh_kf, sh_red, sh_tile);
        if (tid < 16) {
          int i = r0 + tid;
          xcur[i] = sxv[i] + sh_tile[tid] * denv[i];
        }
      }
      grid_barrier(bar);
    }
  }

  if (wg == 0)
    for (int i = tid; i < E; i += NT) out_x[i] = xcur[i];
}

extern "C" void kernel_launch(void* const* d_in, const int* in_sizes, int n_in,
                              void* d_out, int out_size, void* d_ws, size_t ws_size,
                              hipStream_t stream) {
  (void)in_sizes; (void)n_in; (void)out_size; (void)ws_size;
  const float* x = (const float*)d_in[0];
  const float* state = (const float*)d_in[1];
  const float* ln1w = (const float*)d_in[2];
  const float* ln1b = (const float*)d_in[3];
  const float* ln2w = (const float*)d_in[4];
  const float* ln2b = (const float*)d_in[5];
  const float* tdec = (const float*)d_in[6];
  const float* tfirst = (const float*)d_in[7];
  const float* keyw = (const float*)d_in[8];
  const float* outw = (const float*)d_in[9];
  const float* mixk = (const float*)d_in[10];
  const float* mixv = (const float*)d_in[11];
  const float* mixr = (const float*)d_in[12];
  const float* mixkf = (const float*)d_in[13];
  const float* mixrf = (const float*)d_in[14];
  const float* keyffn = (const float*)d_in[15];
  const float* recffn = (const float*)d_in[16];
  const float* valffn = (const float*)d_in[17];

  float* ws = (float*)d_ws;
  unsigned* bar = (unsigned*)(ws + WS_BAR);

  const float* statea = state;
  const float* stateb = state + (size_t)LAYERS * E;
  const float* statec = state + (size_t)2 * LAYERS * E;
  const float* stated = state + (size_t)3 * LAYERS * E;

  rwkv_init_kernel<<<4, 256, 0, stream>>>(x, ws, bar);
  rwkv_fused_kernel<<<NWG, NT, 0, stream>>>(
      statea, stateb, statec, stated, ln1w, ln1b, ln2w, ln2b, tdec, tfirst,
      keyw, outw, mixk, mixv, mixr, mixkf, mixrf, keyffn, recffn, valffn,
      (float*)d_out, ws, bar);
}